// GraphAttention_29489245454920
// MI455X (gfx1250) — compile-verified
//
#include <hip/hip_runtime.h>
#include <hip/hip_bf16.h>

typedef __attribute__((ext_vector_type(2))) float f32x2;
typedef __attribute__((ext_vector_type(8))) float f32x8;

#define WMMA_F32(a, b, c) \
  __builtin_amdgcn_wmma_f32_16x16x4_f32(false, (a), false, (b), (short)0, (c), false, false)

// ---------------------------------------------------------------------------
// CDNA5 async memory->LDS path (ASYNCcnt-tracked, no VGPR roundtrip).
// Builtin signature (from hipcc diagnostic): param0 = int __vector_size__(16)
// in the __device__ (AS1) space, param1 = LDS (AS3), then imm offset + cpol.
// ---------------------------------------------------------------------------
#if __has_builtin(__builtin_amdgcn_global_load_async_to_lds_b128) && \
    __has_builtin(__builtin_amdgcn_s_wait_asynccnt)
#define HAVE_ASYNC_LDS 1
typedef int v4i_raw __attribute__((vector_size(16)));
typedef __attribute__((address_space(1))) v4i_raw gv4i;
typedef __attribute__((address_space(3))) v4i_raw lv4i;
__device__ __forceinline__ void async_g2l_b128(const float* g, float* l) {
  // Generic LDS address carries the LDS byte offset in its low 32 bits
  // (ISA 10.2 aperture rules), so the truncating cast yields a valid p3.
  __builtin_amdgcn_global_load_async_to_lds_b128(
      (gv4i*)(unsigned long long)g,
      (lv4i*)(unsigned int)(unsigned long long)l, 0, 0);
}
__device__ __forceinline__ void async_wait() {
  __builtin_amdgcn_s_wait_asynccnt(0);
}
#else
#define HAVE_ASYNC_LDS 0
#endif

// ---------------------------------------------------------------------------
// Kernel 1: QKV projection.  X:[B*S=4096, 1024] x W^T -> [B,H,S,D] layout.
// grid = (8 col-blocks, 256 row-tiles, 3 matrices), block = 256 (8 waves).
// Each wave: one 16x16 output tile, K=1024 via 256 chained f32 WMMAs.
// ---------------------------------------------------------------------------
__global__ void qkv_proj_kernel(const float* __restrict__ X,
                                const float* __restrict__ Wq, const float* __restrict__ bq,
                                const float* __restrict__ Wk, const float* __restrict__ bk,
                                const float* __restrict__ Wv, const float* __restrict__ bv,
                                float* __restrict__ Qg, float* __restrict__ Kg,
                                float* __restrict__ Vg) {
  const int mat = blockIdx.z;
  const float* W;
  const float* bias;
  float* Out;
  if (mat == 0)      { W = Wq; bias = bq; Out = Qg; }
  else if (mat == 1) { W = Wk; bias = bk; Out = Kg; }
  else               { W = Wv; bias = bv; Out = Vg; }

  const int rbase = blockIdx.y * 16;   // row tile in [0,4096)
  const int cbase = blockIdx.x * 128;  // col block in [0,1024)

  extern __shared__ float smem[];
  const int XS = 1028;  // padded row stride: 1028*4 % 16 == 0 (B128-aligned),
                        // 1028 % 64 == 4 -> rows spread across banks.

  // Stage X tile [16][1024] into LDS (float4 / B128 granularity).
  for (int idx = threadIdx.x; idx < 16 * 256; idx += 256) {
    const int r  = idx >> 8;
    const int c4 = idx & 255;
    const float* src = X + (size_t)(rbase + r) * 1024 + c4 * 4;
    float* dst = smem + r * XS + c4 * 4;
#if HAVE_ASYNC_LDS
    async_g2l_b128(src, dst);
#else
    const float4 v = *(const float4*)src;
    dst[0] = v.x; dst[1] = v.y; dst[2] = v.z; dst[3] = v.w;
#endif
  }
#if HAVE_ASYNC_LDS
  async_wait();
#endif
  __syncthreads();

  const int wave = threadIdx.x >> 5;
  const int lane = threadIdx.x & 31;
  const int half = lane >> 4;
  const int l16  = lane & 15;

  const int jbase = cbase + wave * 16;
  const float* wrow = W + (size_t)(jbase + l16) * 1024;  // B fragment: col n = l16
  const float* arow = smem + l16 * XS;                   // A fragment: row m = l16

  f32x8 acc = {};
  for (int c = 0; c < 256; ++c) {
    const int k0 = c * 4 + half * 2;
    f32x2 a; a.x = arow[k0]; a.y = arow[k0 + 1];
    f32x2 bf; bf.x = wrow[k0]; bf.y = wrow[k0 + 1];
    acc = WMMA_F32(a, bf, acc);
  }

  const int j = jbase + l16;          // output column (n = l16)
  const int h = j >> 6;
  const int d = j & 63;
  const float bj = bias[j];
#pragma unroll
  for (int r = 0; r < 8; ++r) {
    const int m  = r + half * 8;
    const int bs = rbase + m;
    const int bb = bs >> 11;          // batch
    const int ss = bs & 2047;         // sequence
    Out[(((size_t)bb * 16 + h) * 2048 + ss) * 64 + d] = acc[r] + bj;
  }
}

// ---------------------------------------------------------------------------
// Kernel 2: relative-position attention for one (b, h, 16-query tile).
// grid = (128 qtiles, 16 heads, 2 batch), block = 256 (8 waves).
// Dynamic LDS ~178KB (WGP has 320KB): full 16x2048 score row resident.
// ---------------------------------------------------------------------------
__global__ void rel_attn_kernel(const float* __restrict__ Qg,
                                const float* __restrict__ Kg,
                                const float* __restrict__ Vg,
                                const float* __restrict__ st_mask,
                                const float* __restrict__ rel_key,
                                const float* __restrict__ rel_val,
                                float* __restrict__ out) {
  const int qtile = blockIdx.x;
  const int h     = blockIdx.y;
  const int b     = blockIdx.z;
  const int qbase = qtile * 16;
  const size_t bhS = ((size_t)b * 16 + h) * 2048;

  extern __shared__ float smem[];
  const int QS  = 68;                   // B128-aligned padded stride
  float* q_s    = smem;                 // [16][68]
  float* mask_s = q_s + 16 * QS;        // [2048]
  float* rels_s = mask_s + 2048;        // [16][260]  rel_scores
  float* relp_s = rels_s + 16 * 260;    // [16][260]  rel_probs
  float* red_s  = relp_s + 16 * 260;    // [16][16]   partial reductions
  float* row_s  = red_s + 256;          // [32]       rowmax / rowinv
  float* ctxr_s = row_s + 32;           // [16][66]   rel-val context
  float* sc_s   = ctxr_s + 16 * 66;     // [16][2050] scores/probs
  const int SCS = 2050;

  const int tid  = threadIdx.x;
  const int wave = tid >> 5;
  const int lane = tid & 31;
  const int half = lane >> 4;
  const int l16  = lane & 15;

  // Stage Q tile [16][64] (async B128 path when available) and mask row.
  {
    const int r  = tid >> 4;            // 256 threads = 16 rows x 16 float4
    const int c4 = tid & 15;
    const float* src = Qg + (bhS + qbase + r) * 64 + c4 * 4;
    float* dst = q_s + r * QS + c4 * 4;
#if HAVE_ASYNC_LDS
    async_g2l_b128(src, dst);
#else
    const float4 v = *(const float4*)src;
    dst[0] = v.x; dst[1] = v.y; dst[2] = v.z; dst[3] = v.w;
#endif
  }
  for (int k = tid; k < 2048; k += 256)
    mask_s[k] = (1.0f - st_mask[(size_t)b * 2048 + k]) * -10000.0f;
#if HAVE_ASYNC_LDS
  async_wait();
#endif
  __syncthreads();

  const float* qrow = q_s + l16 * QS;   // A fragment row m = l16

  // ---- rel_scores[16][257] = Q . rel_key^T  (17 column tiles over 8 waves)
  for (int t = wave; t < 17; t += 8) {
    const int rcol = t * 16 + l16;
    const int rc   = rcol > 256 ? 256 : rcol;
    const float* krow = rel_key + rc * 64;
    f32x8 acc = {};
#pragma unroll
    for (int c = 0; c < 16; ++c) {
      const int k0 = c * 4 + half * 2;
      f32x2 a;  a.x = qrow[k0];  a.y = qrow[k0 + 1];
      f32x2 bf; bf.x = krow[k0]; bf.y = krow[k0 + 1];
      acc = WMMA_F32(a, bf, acc);
    }
    if (rcol < 257) {
#pragma unroll
      for (int r = 0; r < 8; ++r)
        rels_s[(r + half * 8) * 260 + rcol] = acc[r];
    }
  }
  __syncthreads();

  // ---- scores[16][2048]: Q.K^T / sqrt(D) + rel bias + mask
  const float invs = 0.125f;            // 1/sqrt(64)
  for (int kt = wave; kt < 128; kt += 8) {
    const int kcol = kt * 16 + l16;
    const float* kr = Kg + (bhS + kcol) * 64;
    __builtin_prefetch(Kg + (bhS + ((kt + 8) & 127) * 16 + l16) * 64, 0, 0);
    f32x8 acc = {};
#pragma unroll
    for (int c = 0; c < 16; ++c) {
      const int k0 = c * 4 + half * 2;
      f32x2 a;  a.x = qrow[k0]; a.y = qrow[k0 + 1];
      f32x2 bf; bf.x = kr[k0];  bf.y = kr[k0 + 1];
      acc = WMMA_F32(a, bf, acc);
    }
#pragma unroll
    for (int r = 0; r < 8; ++r) {
      const int m  = r + half * 8;
      const int qg = qbase + m;
      int bu = kcol - qg + 128;
      bu = bu < 0 ? 0 : (bu > 256 ? 256 : bu);
      sc_s[m * SCS + kcol] = acc[r] * invs + rels_s[m * 260 + bu] + mask_s[kcol];
    }
  }
  __syncthreads();

  // ---- softmax (two-pass, rows resident in LDS)
  const int rr  = tid & 15;
  const int seg = tid >> 4;             // 16 segments x 128 cols
  {
    const float* srow = sc_s + rr * SCS + seg * 128;
    float pm = -1e30f;
    for (int k = 0; k < 128; ++k) pm = fmaxf(pm, srow[k]);
    red_s[rr * 16 + seg] = pm;
  }
  __syncthreads();
  if (tid < 16) {
    float pm = -1e30f;
    for (int s2 = 0; s2 < 16; ++s2) pm = fmaxf(pm, red_s[tid * 16 + s2]);
    row_s[tid] = pm;
  }
  __syncthreads();
  {
    const float rm = row_s[rr];
    float* srow = sc_s + rr * SCS + seg * 128;
    float ps = 0.0f;
    for (int k = 0; k < 128; ++k) {
      const float p = __expf(srow[k] - rm);
      srow[k] = p;
      ps += p;
    }
    red_s[rr * 16 + seg] = ps;
  }
  __syncthreads();
  if (tid < 16) {
    float s2 = 0.0f;
    for (int i = 0; i < 16; ++i) s2 += red_s[tid * 16 + i];
    row_s[16 + tid] = 1.0f / s2;
  }
  __syncthreads();
  {
    const float ri = row_s[16 + rr];
    float* srow = sc_s + rr * SCS + seg * 128;
    for (int k = 0; k < 128; ++k) srow[k] *= ri;
  }
  // zero rel_probs, then scatter probs into relative buckets (ds_add_f32)
  for (int i = tid; i < 16 * 260; i += 256) relp_s[i] = 0.0f;
  __syncthreads();
  {
    const float* srow = sc_s + rr * SCS;
    const int qg = qbase + rr;
    for (int k = seg * 128; k < seg * 128 + 128; ++k) {
      int bu = k - qg + 128;
      bu = bu < 0 ? 0 : (bu > 256 ? 256 : bu);
      atomicAdd(&relp_s[rr * 260 + bu], srow[k]);
    }
  }
  __syncthreads();

  // ---- context: waves 0-3 do P.V (K=2048); waves 4-7 do rel_probs.rel_val (K=260)
  f32x8 acc = {};
  int d;
  if (wave < 4) {
    d = wave * 16 + l16;                // B fragment col n = l16
    const float* arow = sc_s + l16 * SCS;
    const float* vb = Vg + bhS * 64 + d;
    for (int kc = 0; kc < 512; ++kc) {
      const int k0 = kc * 4 + half * 2;
      f32x2 a;  a.x = arow[k0]; a.y = arow[k0 + 1];
      f32x2 bf; bf.x = vb[(size_t)k0 * 64]; bf.y = vb[(size_t)(k0 + 1) * 64];
      acc = WMMA_F32(a, bf, acc);
    }
  } else {
    d = (wave - 4) * 16 + l16;
    const float* arow = relp_s + l16 * 260;
    for (int kc = 0; kc < 65; ++kc) {
      const int k0 = kc * 4 + half * 2;
      const int r0 = k0 > 256 ? 256 : k0;          // rows 257..259 have zero A
      const int r1 = (k0 + 1) > 256 ? 256 : (k0 + 1);
      f32x2 a;  a.x = arow[k0];              a.y = arow[k0 + 1];
      f32x2 bf; bf.x = rel_val[r0 * 64 + d]; bf.y = rel_val[r1 * 64 + d];
      acc = WMMA_F32(a, bf, acc);
    }
#pragma unroll
    for (int r = 0; r < 8; ++r)
      ctxr_s[(r + half * 8) * 66 + d] = acc[r];
  }
  __syncthreads();

  if (wave < 4) {
#pragma unroll
    for (int r = 0; r < 8; ++r) {
      const int m = r + half * 8;
      const float y = acc[r] + ctxr_s[m * 66 + d];
      out[((size_t)b * 2048 + qbase + m) * 1024 + h * 64 + d] = y;
    }
  }
}

// ---------------------------------------------------------------------------
extern "C" void kernel_launch(void* const* d_in, const int* in_sizes, int n_in,
                              void* d_out, int out_size, void* d_ws, size_t ws_size,
                              hipStream_t stream) {
  const float* X  = (const float*)d_in[0];   // hidden_states [2,2048,1024]
  const float* sm = (const float*)d_in[1];   // st_mask [2,2048]
  const float* Wq = (const float*)d_in[2];
  const float* bq = (const float*)d_in[3];
  const float* Wk = (const float*)d_in[4];
  const float* bk = (const float*)d_in[5];
  const float* Wv = (const float*)d_in[6];
  const float* bv = (const float*)d_in[7];
  const float* rk = (const float*)d_in[8];   // rel_key_emb [257,64]
  const float* rv = (const float*)d_in[9];   // rel_val_emb [257,64]
  // d_in[10] = indices — recomputed analytically on device, unused.
  float* out = (float*)d_out;

  // Workspace: Q,K,V in [B,H,S,D], 4,194,304 floats each (48 MB total).
  float* Qg = (float*)d_ws;
  float* Kg = Qg + 4194304;
  float* Vg = Kg + 4194304;

  const size_t smem1 = (size_t)16 * 1028 * sizeof(float);               // ~64 KB
  qkv_proj_kernel<<<dim3(8, 256, 3), 256, smem1, stream>>>(
      X, Wq, bq, Wk, bk, Wv, bv, Qg, Kg, Vg);

  const size_t smem2 =
      (size_t)(16 * 68 + 2048 + 16 * 260 + 16 * 260 + 256 + 32 + 16 * 66 +
               16 * 2050) * sizeof(float);                              // ~178 KB
  rel_attn_kernel<<<dim3(128, 16, 2), 256, smem2, stream>>>(
      Qg, Kg, Vg, sm, rk, rv, out);
}